// KGATLayer_49383533970017
// MI455X (gfx1250) — compile-verified
//
#include <hip/hip_runtime.h>
#include <hip/hip_bf16.h>

typedef __attribute__((ext_vector_type(2))) float v2f;
typedef __attribute__((ext_vector_type(8))) float v8f;

#define NEG_SLOPE 0.2f

// ---------------------------------------------------------------------------
// C(M x Nc) = A(M x K, row-major) * W^T   where W is (Nc x K, row-major).
// One 16x16 output tile per wave, 4 waves per block cover Nc=64 columns.
// K stepped by 4 with V_WMMA_F32_16X16X4_F32 (full f32 precision).
// ---------------------------------------------------------------------------
__global__ void wmma_gemm_bt(const float* __restrict__ A,
                             const float* __restrict__ W,
                             float* __restrict__ C,
                             int M, int K, int Nc) {
    const int wave   = threadIdx.x >> 5;
    const int lane   = threadIdx.x & 31;
    const int half   = lane >> 4;     // 0: lanes 0-15, 1: lanes 16-31
    const int lanelo = lane & 15;
    const int row0   = blockIdx.x * 16;
    const int col0   = wave * 16;
    if (col0 >= Nc) return;                       // wave-uniform

    // clamp A row for ragged M (rel GEMM); lane-uniform data path, no branch
    int arow = row0 + lanelo;
    if (arow >= M) arow = M - 1;
    const int brow = col0 + lanelo;               // always < Nc by grid

    const float* __restrict__ arp = A + (size_t)arow * K;
    const float* __restrict__ brp = W + (size_t)brow * K;

    v8f c = {};
    for (int k0 = 0; k0 < K; k0 += 4) {
        // A frag: lane holds A[arow, k0+2*half .. +1]
        v2f a = *(const v2f*)(arp + k0 + 2 * half);
        // B frag: B[k, n] = W[n, k]; lane holds W[brow, k0+2*half .. +1]
        v2f b = *(const v2f*)(brp + k0 + 2 * half);
        c = __builtin_amdgcn_wmma_f32_16x16x4_f32(
                false, a, false, b, (short)0, c, false, false);
    }
#pragma unroll
    for (int v = 0; v < 8; ++v) {
        int r = row0 + v + 8 * half;              // C/D layout: VGPR v -> rows v, v+8
        if (r < M) C[(size_t)r * Nc + col0 + lanelo] = c[v];
    }
}

// ---------------------------------------------------------------------------
// Per-node attention scalars + init of segment-max / denom accumulators.
// ---------------------------------------------------------------------------
__global__ void node_scores(const float* __restrict__ Wh,
                            const float* __restrict__ att,   // attvec[0], 3D floats
                            float* __restrict__ s1, float* __restrict__ s2,
                            float* __restrict__ emax, float* __restrict__ denom,
                            int N, int D) {
    int n = blockIdx.x * blockDim.x + threadIdx.x;
    if (n >= N) return;
    const float* row = Wh + (size_t)n * D;
    float a1 = 0.f, a2 = 0.f;
    for (int k = 0; k < D; k += 4) {
        float4 w = *(const float4*)(row + k);
        a1 += w.x * att[k]     + w.y * att[k + 1]     + w.z * att[k + 2]     + w.w * att[k + 3];
        a2 += w.x * att[D + k] + w.y * att[D + k + 1] + w.z * att[D + k + 2] + w.w * att[D + k + 3];
    }
    s1[n] = a1;
    s2[n] = a2;
    emax[n]  = __int_as_float(0xFF800000);   // -inf
    denom[n] = 0.f;
}

// Per-relation attention scalar: srel[r] = Wrel_all[r] . att[2D:]
__global__ void rel_scores(const float* __restrict__ Wrel,
                           const float* __restrict__ att2D,
                           float* __restrict__ srel, int R, int D) {
    int r = blockIdx.x * blockDim.x + threadIdx.x;
    if (r >= R) return;
    const float* row = Wrel + (size_t)r * D;
    float a = 0.f;
    for (int k = 0; k < D; ++k) a += row[k] * att2D[k];
    srel[r] = a;
}

__global__ void zero_f32(float* __restrict__ p, size_t n) {
    size_t i = (size_t)blockIdx.x * blockDim.x + threadIdx.x;
    if (i < n) p[i] = 0.f;
}

// ---------------------------------------------------------------------------
// Edge pass 1: logits + leaky-relu + segment max (bit-trick float atomic max).
// ---------------------------------------------------------------------------
__global__ void edge_logits(const int* __restrict__ src, const int* __restrict__ dst,
                            const int* __restrict__ rel,
                            const float* __restrict__ s1, const float* __restrict__ s2,
                            const float* __restrict__ srel,
                            float* __restrict__ earr, float* __restrict__ emax, int E) {
    int e = blockIdx.x * blockDim.x + threadIdx.x;
    if (e >= E) return;
    int s = src[e];
    float v = s1[s] + s2[dst[e]] + srel[rel[e]];
    v = v > 0.f ? v : NEG_SLOPE * v;
    earr[e] = v;
    // monotone float-max via signed-int max (v>=0) / unsigned-int min (v<0)
    if (v >= 0.f) atomicMax((int*)(emax + s), __float_as_int(v));
    else          atomicMin((unsigned int*)(emax + s), __float_as_uint(v));
}

// Edge pass 2: numerator + denominator segment sum.
__global__ void edge_exp(const int* __restrict__ src,
                         float* __restrict__ earr,
                         const float* __restrict__ emax,
                         float* __restrict__ denom, int E) {
    int e = blockIdx.x * blockDim.x + threadIdx.x;
    if (e >= E) return;
    int s = src[e];
    float num = __expf(earr[e] - emax[s]);
    earr[e] = num;
    __hip_atomic_fetch_add(denom + s, num, __ATOMIC_RELAXED, __HIP_MEMORY_SCOPE_AGENT);
}

// ---------------------------------------------------------------------------
// Edge pass 3: alpha * (Wh[dst] + Wrel[rel]) scattered into H_out[src].
// One wave per edge; lane owns 2 contiguous columns (D==64) -> coalesced 8B.
// ---------------------------------------------------------------------------
__global__ void edge_scatter(const int* __restrict__ src, const int* __restrict__ dst,
                             const int* __restrict__ rel,
                             const float* __restrict__ Wh, const float* __restrict__ Wrel,
                             const float* __restrict__ earr, const float* __restrict__ denom,
                             float* __restrict__ out, int E, int D) {
    const int wave = threadIdx.x >> 5;
    const int lane = threadIdx.x & 31;
    const int e = blockIdx.x * 8 + wave;
    if (e >= E) return;
    const int s = src[e], d = dst[e], r = rel[e];
    const float alpha = earr[e] / (denom[s] + 1e-12f);
    const float2 h = *(const float2*)(Wh   + (size_t)d * D + lane * 2);
    const float2 w = *(const float2*)(Wrel + (size_t)r * D + lane * 2);
    float m0 = alpha * (h.x + w.x);
    float m1 = alpha * (h.y + w.y);
    float* p = out + (size_t)s * D + lane * 2;
    __hip_atomic_fetch_add(p,     m0, __ATOMIC_RELAXED, __HIP_MEMORY_SCOPE_AGENT);
    __hip_atomic_fetch_add(p + 1, m1, __ATOMIC_RELAXED, __HIP_MEMORY_SCOPE_AGENT);
}

extern "C" void kernel_launch(void* const* d_in, const int* in_sizes, int n_in,
                              void* d_out, int out_size, void* d_ws, size_t ws_size,
                              hipStream_t stream) {
    const float* H       = (const float*)d_in[0];
    const float* W_node  = (const float*)d_in[1];
    const float* W_rel   = (const float*)d_in[2];
    const float* attvec  = (const float*)d_in[3];
    const float* rel_emb = (const float*)d_in[4];
    const int*   src     = (const int*)d_in[5];
    const int*   dst     = (const int*)d_in[6];
    const int*   rel     = (const int*)d_in[7];

    const int D  = 64;
    const int RD = in_sizes[2] / D;                 // W_rel is (D, RD)
    const int R  = in_sizes[4] / RD;                // NUM_RELS
    const int N  = in_sizes[0] / D;
    const int E  = in_sizes[5];

    float* ws    = (float*)d_ws;
    float* Wh    = ws;  ws += (size_t)N * D;
    float* Wrel  = ws;  ws += (size_t)R * D;
    float* s1    = ws;  ws += N;
    float* s2    = ws;  ws += N;
    float* srel  = ws;  ws += R;
    float* earr  = ws;  ws += E;
    float* emax  = ws;  ws += N;
    float* denom = ws;  ws += N;
    float* out   = (float*)d_out;

    // H_out accumulator must be zero each call (atomics)
    {
        size_t tot = (size_t)N * D;
        zero_f32<<<dim3((unsigned)((tot + 255) / 256)), 256, 0, stream>>>(out, tot);
    }
    // Wh = H @ W_node^T          (WMMA f32)
    wmma_gemm_bt<<<dim3((N + 15) / 16), 128, 0, stream>>>(H, W_node, Wh, N, D, D);
    // Wrel_all = rel_emb @ W_rel^T (WMMA f32, ragged rows)
    wmma_gemm_bt<<<dim3((R + 15) / 16), 128, 0, stream>>>(rel_emb, W_rel, Wrel, R, RD, D);
    // per-node / per-relation attention scalars + accumulator init
    node_scores<<<dim3((N + 255) / 256), 256, 0, stream>>>(Wh, attvec, s1, s2, emax, denom, N, D);
    rel_scores <<<dim3((R + 255) / 256), 256, 0, stream>>>(Wrel, attvec + 2 * D, srel, R, D);
    // segment softmax over edges grouped by src
    edge_logits<<<dim3((E + 255) / 256), 256, 0, stream>>>(src, dst, rel, s1, s2, srel, earr, emax, E);
    edge_exp   <<<dim3((E + 255) / 256), 256, 0, stream>>>(src, earr, emax, denom, E);
    // weighted message scatter
    edge_scatter<<<dim3((E + 7) / 8), 256, 0, stream>>>(src, dst, rel, Wh, Wrel, earr, denom, out, E, D);
}